// EdgeLayer_47382079209911
// MI455X (gfx1250) — compile-verified
//
#include <hip/hip_runtime.h>

#define BB 4
#define NN 1024
#define DD 256
#define CC 8
#define KNB 16
#define SLOTS 17
#define QKW 4096            // 2*C*D per (b,n) row of qk
#define SCALE 0.0625f       // D^-0.5 = 1/16
#define DCH 32              // d-chunk staged per TDM transfer

typedef float v2f __attribute__((ext_vector_type(2)));
typedef float v8f __attribute__((ext_vector_type(8)));
typedef unsigned int u32x4 __attribute__((ext_vector_type(4)));
typedef int i32x4 __attribute__((ext_vector_type(4)));
typedef int i32x8 __attribute__((ext_vector_type(8)));

__device__ __forceinline__ v8f wmma_f32_4(v2f a, v2f b, v8f c) {
  // D = A(16x4,f32) * B(4x16,f32) + C(16x16,f32)
  return __builtin_amdgcn_wmma_f32_16x16x4_f32(false, a, false, b, (short)0, c,
                                               false, false);
}

__device__ __forceinline__ v8f v8f_zero() {
  v8f z = {0.f, 0.f, 0.f, 0.f, 0.f, 0.f, 0.f, 0.f};
  return z;
}

// TDM: stage a 2D panel [rows x DCH f32], row stride in elements, into LDS.
// D# layout per CDNA5 ISA §8 (group0 128b, group1 256b; groups 2/3 unused=0).
// This toolchain exposes the 6-arg clang-23 builtin form.
__device__ __forceinline__ void tdm_load_2d(const float* gaddr, unsigned lds_off,
                                            int rows, int rowstride_elems) {
  unsigned long long ga = (unsigned long long)(uintptr_t)gaddr;
  u32x4 g0;
  g0.x = 1u;                                         // count=1, user mode
  g0.y = lds_off;                                    // lds_addr (bytes)
  g0.z = (unsigned)ga;                               // global_addr[31:0]
  g0.w = (unsigned)((ga >> 32) & 0x1FFFFFFu) | (2u << 30);  // addr[56:32], type=2
  i32x8 g1;
  g1[0] = 0x00020000;                 // workgroup_mask=0, data_size=2 (4B)
  g1[1] = (int)((unsigned)DCH << 16); // tensor_dim0[15:0] in bits 63:48
  g1[2] = (int)((unsigned)rows << 16);// tensor_dim0 hi=0 | tensor_dim1[15:0]
  g1[3] = (int)((unsigned)DCH << 16); // tensor_dim1 hi=0 | tile_dim0
  g1[4] = rows;                       // tile_dim1 | tile_dim2=0
  g1[5] = rowstride_elems;            // tensor_dim0_stride[31:0]
  g1[6] = 0;                          // stride hi | dim1_stride lo (unused, 2D)
  g1[7] = 0;
  i32x4 z4 = {0, 0, 0, 0};
  i32x8 z8 = {0, 0, 0, 0, 0, 0, 0, 0};
  __builtin_amdgcn_tensor_load_to_lds(g0, g1, z4, z4, z8, 0);
}

// ---------------------------------------------------------------------------
// Kernel 1: qk = x @ W_qk   (M=4096, K=256, N=4096), f32 WMMA 16x16x4
// Block: 256 thr = 8 waves; wave computes 16x64; block tile 32x256.
// ---------------------------------------------------------------------------
__global__ __launch_bounds__(256) void k_gemm_qk(const float* __restrict__ x,
                                                 const float* __restrict__ W,
                                                 float* __restrict__ qk) {
  const int lane = threadIdx.x & 31;
  const int wave = threadIdx.x >> 5;
  const int half = lane >> 4;
  const int l15  = lane & 15;
  const int m0 = blockIdx.y * 32 + (wave >> 2) * 16;
  const int n0 = blockIdx.x * 256 + (wave & 3) * 64;

  v8f acc[4];
#pragma unroll
  for (int i = 0; i < 4; ++i) acc[i] = v8f_zero();

  const float* xrow = x + (size_t)(m0 + l15) * DD;
  for (int kk = 0; kk < DD; kk += 4) {
    v2f a = *(const v2f*)(xrow + kk + 2 * half);  // A[m, kk..kk+3] striped
#pragma unroll
    for (int i = 0; i < 4; ++i) {
      int n = n0 + i * 16 + l15;
      v2f b;
      b.x = W[(size_t)(kk + 2 * half) * QKW + n];
      b.y = W[(size_t)(kk + 2 * half + 1) * QKW + n];
      acc[i] = wmma_f32_4(a, b, acc[i]);
    }
  }
#pragma unroll
  for (int i = 0; i < 4; ++i) {
#pragma unroll
    for (int r = 0; r < 8; ++r) {
      int row = m0 + r + 8 * half;
      int col = n0 + i * 16 + l15;
      qk[(size_t)row * QKW + col] = acc[i][r];
    }
  }
}

__global__ void k_zero(float* __restrict__ p, int n) {
  int t = blockIdx.x * blockDim.x + threadIdx.x;
  if (t < n) p[t] = 0.f;
}

// ---------------------------------------------------------------------------
// Kernel 2: fused per (b, 16-row n-tile):
//   per channel: TDM-stage K panel (1024 x 32 f32) into LDS chunk-by-chunk,
//   S = scale*Q.K^T via WMMA (B-frags from LDS) -> softmax stats (row max, Z)
//   -> channel-summed probs in LDS -> top-16 + diag.
// Dynamic LDS: s_tile 64KB + sum_tile 64KB + k_panel 128KB = 256KB (WGP 320KB).
// ---------------------------------------------------------------------------
__global__ __launch_bounds__(256) void k_attn_stats(
    const float* __restrict__ qk, float* __restrict__ maxv,
    float* __restrict__ zv, int* __restrict__ idxb, int* __restrict__ cntb) {
  extern __shared__ float smem[];
  float* s_tile   = smem;                // 16*1024
  float* sum_tile = smem + 16 * NN;      // 16*1024
  float* k_lds    = smem + 32 * NN;      // 1024*DCH
  __shared__ int diagf[16];

  const int b = blockIdx.y;
  const int n0 = blockIdx.x * 16;
  const int lane = threadIdx.x & 31;
  const int wave = threadIdx.x >> 5;
  const int half = lane >> 4;
  const int l15  = lane & 15;
  const int wm0  = wave * 128;  // wave's 128-column strip of [0,1024)

  if (threadIdx.x < 16) diagf[threadIdx.x] = 0;

  const size_t qkb = (size_t)b * NN * QKW;
  const unsigned klds_off = (unsigned)(uintptr_t)k_lds;  // LDS byte offset

  for (int c = 0; c < CC; ++c) {
    v8f acc[8];
#pragma unroll
    for (int i = 0; i < 8; ++i) acc[i] = v8f_zero();

    const float* qrow = qk + qkb + (size_t)(n0 + l15) * QKW + c * DD;

    for (int dt = 0; dt < DD / DCH; ++dt) {
      // wave 0 DMAs K[0..1023, c*256+dt*32 .. +32) into LDS (TENSORcnt)
      if (wave == 0) {
        const float* kpanel = qk + qkb + 2048 + c * DD + dt * DCH;
        tdm_load_2d(kpanel, klds_off, NN, QKW);
        __builtin_amdgcn_s_wait_tensorcnt(0);
      }
      __syncthreads();  // publish panel

      const float* qchunk = qrow + dt * DCH;
#pragma unroll
      for (int kl = 0; kl < DCH; kl += 4) {
        v2f a = *(const v2f*)(qchunk + kl + 2 * half);
#pragma unroll
        for (int i = 0; i < 8; ++i) {
          int m = wm0 + i * 16 + l15;
          v2f bf = *(const v2f*)(k_lds + m * DCH + kl + 2 * half);
          acc[i] = wmma_f32_4(a, bf, acc[i]);
        }
      }
      __syncthreads();  // all reads done before next DMA overwrites panel
    }

    // store scaled S tile to LDS (C/D layout: VGPR r -> M=r / M=r+8)
#pragma unroll
    for (int i = 0; i < 8; ++i)
#pragma unroll
      for (int r = 0; r < 8; ++r)
        s_tile[(r + 8 * half) * NN + wm0 + i * 16 + l15] = acc[i][r] * SCALE;
    __syncthreads();

    // softmax over each row; wave w owns rows 2w, 2w+1
    for (int rr = 0; rr < 2; ++rr) {
      int row = wave * 2 + rr;
      float mx = -3.4e38f;
      for (int s = lane; s < NN; s += 32) mx = fmaxf(mx, s_tile[row * NN + s]);
      for (int off = 16; off > 0; off >>= 1) mx = fmaxf(mx, __shfl_down(mx, off));
      mx = __shfl(mx, 0);
      float zs = 0.f;
      for (int s = lane; s < NN; s += 32) zs += __expf(s_tile[row * NN + s] - mx);
      for (int off = 16; off > 0; off >>= 1) zs += __shfl_down(zs, off);
      zs = __shfl(zs, 0);
      float inv = 1.f / zs;
      for (int s = lane; s < NN; s += 32) {
        float p = __expf(s_tile[row * NN + s] - mx) * inv;
        if (c == 0) sum_tile[row * NN + s] = p;
        else        sum_tile[row * NN + s] += p;
      }
      if (lane == 0) {
        maxv[((size_t)(b * CC + c)) * NN + n0 + row] = mx;
        zv  [((size_t)(b * CC + c)) * NN + n0 + row] = zs;
      }
    }
    __syncthreads();
  }

  // top-16 per row of channel-summed probs (iterative argmax, wave per row)
  for (int rr = 0; rr < 2; ++rr) {
    int row = wave * 2 + rr;
    int ng  = n0 + row;
    for (int it = 0; it < KNB; ++it) {
      float bv = -1.f;
      int   bi = 0;
      for (int s = lane; s < NN; s += 32) {
        float v = sum_tile[row * NN + s];
        if (v > bv) { bv = v; bi = s; }
      }
      for (int off = 16; off > 0; off >>= 1) {
        float ov = __shfl_down(bv, off);
        int   oi = __shfl_down(bi, off);
        if (ov > bv || (ov == bv && oi < bi)) { bv = ov; bi = oi; }
      }
      bi = __shfl(bi, 0);
      if (lane == 0) {
        sum_tile[row * NN + bi] = -2.f;  // exclude
        idxb[((size_t)b * SLOTS + it) * NN + ng] = bi;
        if (bi == ng) diagf[row] = 1;
      }
      __syncthreads();
    }
  }
  __syncthreads();
  if (threadIdx.x < 16) {
    int row = threadIdx.x;
    int ng  = n0 + row;
    idxb[((size_t)b * SLOTS + 16) * NN + ng] = ng;   // diagonal slot
    cntb[b * NN + ng] = diagf[row] ? KNB : SLOTS;
  }
}

// ---------------------------------------------------------------------------
// Kernel 3: recompute the <=17 masked attn values per (b,c,n) from qk + (max,Z),
// row-normalize -> nrv [B,C,17,N]; scatter-add column sums (f32 global atomics).
// One wave per (b,c,n); lane j handles slot j.
// ---------------------------------------------------------------------------
__global__ __launch_bounds__(256) void k_compact_nr(
    const float* __restrict__ qk, const float* __restrict__ maxv,
    const float* __restrict__ zv, const int* __restrict__ idxb,
    const int* __restrict__ cntb, float* __restrict__ nrv,
    float* __restrict__ colsum) {
  int gw = blockIdx.x * 8 + (threadIdx.x >> 5);  // 0 .. B*C*N-1
  int lane = threadIdx.x & 31;
  int n = gw & (NN - 1);
  int c = (gw >> 10) & (CC - 1);
  int b = gw >> 13;
  int cnt = cntb[b * NN + n];

  float ev = 0.f;
  int kidx = 0;
  if (lane < SLOTS) {
    kidx = idxb[((size_t)b * SLOTS + lane) * NN + n];
    if (lane < cnt) {
      const float* qrow = qk + ((size_t)(b * NN + n)) * QKW + c * DD;
      const float* krow = qk + ((size_t)(b * NN + kidx)) * QKW + 2048 + c * DD;
      float s = 0.f;
      for (int d = 0; d < DD; d += 4) {
        float4 qv = *(const float4*)(qrow + d);
        float4 kv = *(const float4*)(krow + d);
        s += qv.x * kv.x + qv.y * kv.y + qv.z * kv.z + qv.w * kv.w;
      }
      float mx = maxv[((size_t)(b * CC + c)) * NN + n];
      float Z  = zv  [((size_t)(b * CC + c)) * NN + n];
      ev = __expf(s * SCALE - mx) / Z;
    }
  }
  float rs = ev;
  for (int off = 16; off > 0; off >>= 1) rs += __shfl_down(rs, off);
  rs = __shfl(rs, 0);
  float nr = ev / (rs + 1e-6f);
  if (lane < SLOTS) {
    nrv[(((size_t)(b * CC + c)) * SLOTS + lane) * NN + n] = nr;
    if (lane < cnt)
      atomicAdd(&colsum[((size_t)(b * CC + c)) * NN + kidx], nr);
  }
}

// ---------------------------------------------------------------------------
// Kernel 4: ncv = nrv / (colsum[idx] + 1e-6), elementwise over [B,C,17,N].
// ---------------------------------------------------------------------------
__global__ void k_nc(const float* __restrict__ nrv,
                     const float* __restrict__ colsum,
                     const int* __restrict__ idxb, float* __restrict__ ncv) {
  int t = blockIdx.x * blockDim.x + threadIdx.x;
  if (t >= BB * CC * SLOTS * NN) return;
  int n  = t & (NN - 1);
  int j  = (t >> 10) % SLOTS;
  int bc = t / (NN * SLOTS);
  int b  = bc >> 3;
  int k  = idxb[((size_t)b * SLOTS + j) * NN + n];
  ncv[t] = nrv[t] / (colsum[(size_t)bc * NN + k] + 1e-6f);
}

// ---------------------------------------------------------------------------
// Kernel 5: out[b,c,n,m] = sum_k nr[n,k]*nc[m,k].  Densify 16 nr-rows in LDS
// (ds_add_f32 scatter), then 17-term sparse gather-FMA per output, coalesced
// 134MB store along m.
// ---------------------------------------------------------------------------
__global__ __launch_bounds__(256) void k_final(const float* __restrict__ nrv,
                                               const float* __restrict__ ncv,
                                               const int* __restrict__ idxb,
                                               float* __restrict__ out) {
  extern __shared__ float dense[];  // 16*1024 f32
  const int b  = blockIdx.z;
  const int c  = blockIdx.y;
  const int n0 = blockIdx.x * 16;
  const int tid = threadIdx.x;

  for (int s = tid; s < 16 * NN; s += 256) dense[s] = 0.f;
  __syncthreads();
  for (int jj = tid; jj < 16 * SLOTS; jj += 256) {
    int r = jj / SLOTS, j = jj % SLOTS;
    int n = n0 + r;
    int k = idxb[((size_t)b * SLOTS + j) * NN + n];
    float v = nrv[(((size_t)(b * CC + c)) * SLOTS + j) * NN + n];
    atomicAdd(&dense[r * NN + k], v);  // slot16 may duplicate a top-16 k (val 0)
  }
  __syncthreads();

  const size_t ob = ((size_t)(b * CC + c)) * NN;
#pragma unroll 1
  for (int mi = 0; mi < 4; ++mi) {
    int m = tid + mi * 256;
    int kk[SLOTS];
    float vv[SLOTS];
#pragma unroll
    for (int j = 0; j < SLOTS; ++j) {
      kk[j] = idxb[((size_t)b * SLOTS + j) * NN + m];
      vv[j] = ncv[(((size_t)(b * CC + c)) * SLOTS + j) * NN + m];
    }
#pragma unroll 4
    for (int r = 0; r < 16; ++r) {
      float acc = 0.f;
#pragma unroll
      for (int j = 0; j < SLOTS; ++j) acc += vv[j] * dense[r * NN + kk[j]];
      out[(ob + (n0 + r)) * NN + m] = acc;
    }
  }
}

// ---------------------------------------------------------------------------
extern "C" void kernel_launch(void* const* d_in, const int* in_sizes, int n_in,
                              void* d_out, int out_size, void* d_ws,
                              size_t ws_size, hipStream_t stream) {
  const float* x = (const float*)d_in[0];   // [4,1024,256]
  const float* W = (const float*)d_in[1];   // [256,4096]
  float* out = (float*)d_out;               // [4,8,1024,1024]

  float* ws     = (float*)d_ws;
  float* qk     = ws;                          // 16,777,216 f32 (67MB)
  float* maxv   = qk + (size_t)16777216;       // 32768
  float* zv     = maxv + 32768;                // 32768
  float* nrv    = zv + 32768;                  // 557,056  [B,C,17,N]
  float* ncv    = nrv + 557056;                // 557,056
  float* colsum = ncv + 557056;                // 32768    [B,C,N]
  int*   idxb   = (int*)(colsum + 32768);      // 69,632   [B,17,N]
  int*   cntb   = idxb + 69632;                // 4096     [B,N]

  dim3 g1(16, 128);
  k_gemm_qk<<<g1, 256, 0, stream>>>(x, W, qk);

  k_zero<<<128, 256, 0, stream>>>(colsum, BB * CC * NN);

  dim3 g2(64, 4);
  size_t lds2 = (size_t)(32 * NN + NN * DCH) * sizeof(float);  // 256KB
  k_attn_stats<<<g2, 256, lds2, stream>>>(qk, maxv, zv, idxb, cntb);

  k_compact_nr<<<4096, 256, 0, stream>>>(qk, maxv, zv, idxb, cntb, nrv, colsum);

  k_nc<<<(BB * CC * SLOTS * NN + 255) / 256, 256, 0, stream>>>(nrv, colsum,
                                                               idxb, ncv);

  dim3 g5(64, 8, 4);
  k_final<<<g5, 256, (size_t)16 * NN * sizeof(float), stream>>>(nrv, ncv, idxb,
                                                                out);
}